// SmithWatermanLoss_48541720379546
// MI455X (gfx1250) — compile-verified
//
#include <hip/hip_runtime.h>
#include <cstdint>

// ---------------------------------------------------------------------------
// Smith-Waterman soft (logsumexp) forward on MI455X (gfx1250).
//   x: 2048x2048 f32.  A=B=2047 -> n=4093 anti-diagonals, m=2047 cells each.
//   Latency-bound wavefront DP: one workgroup (1024 thr = 32 wave32 waves on
//   one WGP), 2 cells/thread, DP state in VGPRs, one LDS neighbor exchange +
//   one s_barrier per step, score diagonal streamed one step ahead with
//   gfx1250 async-to-LDS per-lane gathers (ASYNCcnt / s_wait_asynccnt).
//   Whole DP runs in the log2 domain (v_exp_f32/v_log_f32 are base-2 native);
//   diamond-invalid cells (exactly -1e30 in fp32 reference arithmetic) are
//   skipped; time loop unrolled x2 so diagonal parity is compile-time and the
//   h(t-2)<-h(t-1)<-h(t) generation shift is pure register renaming (the step
//   writes h(t) over the dead h(t-2) set; callers swap the two sets).
// ---------------------------------------------------------------------------

#define LOG2E 1.4426950408889634f
#define LN2   0.6931471805599453f
#define NEG2  (-1.4426950408889634e30f)   /* -1e30 * LOG2E */
#define GO2   (-7.213475204444817f)       /* -5  * LOG2E */
#define GE2   (-1.4426950408889634f)      /* -1  * LOG2E */

constexpr int NX = 2048;   // x is NX x NX
constexpr int MC = 2047;   // m: cells per diagonal
constexpr int ND = 4093;   // n: number of diagonals

// gfx1250 async gather: one f32 from global -> LDS, per-lane addresses.
__device__ __forceinline__ void async_gather_f32(const float* g, float* lds) {
  unsigned int       l  = (unsigned int)(uintptr_t)lds;        // LDS byte offset
  unsigned long long ga = (unsigned long long)(uintptr_t)g;    // global address
  asm volatile("global_load_async_to_lds_b32 %0, %1, off"
               :: "v"(l), "v"(ga) : "memory");
}
#define WAIT_ASYNCCNT(N) asm volatile("s_wait_asynccnt %0" :: "i"(N) : "memory")

__device__ __forceinline__ float ex2(float v) {
#if __has_builtin(__builtin_amdgcn_exp2f)
  return __builtin_amdgcn_exp2f(v);      // raw v_exp_f32
#else
  return exp2f(v);
#endif
}
__device__ __forceinline__ float lg2(float v) {
#if __has_builtin(__builtin_amdgcn_logf)
  return __builtin_amdgcn_logf(v);       // raw v_log_f32 (log2)
#else
  return log2f(v);
#endif
}

__device__ __forceinline__ float lse2_2(float a, float b) {
  float m = fmaxf(a, b);
  return m + lg2(ex2(a - m) + ex2(b - m));
}
__device__ __forceinline__ float lse3_2(float a, float b, float c) {
  float m = fmaxf(fmaxf(a, b), c);
  return m + lg2(ex2(a - m) + ex2(b - m) + ex2(c - m));
}
// lse(a,b,c,0): "align" pad appends a literal 0 state.
__device__ __forceinline__ float lse4z_2(float a, float b, float c) {
  float m = fmaxf(fmaxf(a, b), fmaxf(c, 0.0f));
  return m + lg2(ex2(a - m) + ex2(b - m) + ex2(c - m) + ex2(-m));
}
// Online (max, sum2) logsumexp accumulator merge of three values.
__device__ __forceinline__ void merge3_2(float v0, float v1, float v2,
                                         float& rmx, float& rsm) {
  float cm = fmaxf(fmaxf(v0, v1), v2);
  float cs = ex2(v0 - cm) + ex2(v1 - cm) + ex2(v2 - cm);
  if (cm <= rmx) {
    rsm += cs * ex2(cm - rmx);
  } else {
    rsm = rsm * ex2(rmx - cm) + cs;      // rmx may be -3e38 sentinel: ex2->0
    rmx = cm;
  }
}

struct HPair { float a[3], b[3]; };      // one diagonal's states, cells j0/j1
struct Acc {
  float rmx, rsm;                        // online logsumexp accumulator (base-2)
  int   L0, L1;                          // raw linear x-index of diag t+1 gather
};

// One wavefront step. h1 = h(t-1), h2 = h(t-2); writes h(t) OVER h2 (dead
// after this step) so the generation shift is renaming at the call sites.
template <bool EVEN>
__device__ __forceinline__ void sw_step(
    int t, int j0, bool has1, const float* __restrict__ x,
    float (*hbuf)[MC * 3], float (*gbuf)[NX],
    int tlo0, int span0, int tlo1, int span1,
    HPair& h1, HPair& h2, Acc& s)
{
  constexpr int cur = EVEN ? 0 : 1;   // t&1
  constexpr int nxt = EVEN ? 1 : 0;
  const int j1 = j0 + 1;

  // Async prefetch of diagonal t+1 (clamped raw linear index; invalid cells
  // masked on consume). Address advances +1 / +NX alternating per diagonal.
  {
    int c0 = min(max(s.L0, 0), NX * NX - 1);
    int c1 = min(max(s.L1, 0), NX * NX - 1);
    async_gather_f32(x + c0, &gbuf[nxt][j0]);
    async_gather_f32(x + c1, &gbuf[nxt][j1]);
    constexpr int adv = EVEN ? NX : 1;   // diag t+1 -> t+2
    s.L0 += adv;
    s.L1 += adv;
  }
  WAIT_ASYNCCNT(2);   // oldest 2 (diagonal t, issued last step) complete

  // One neighbor cell of h(t-1): j0-1 (even) / j1+1 (odd); clamped + masked.
  float nb0, nb1, nb2;
  {
    const float* hb = hbuf[nxt];        // (t-1)&1 == nxt
    const int jn = EVEN ? (j0 - 1) : (j1 + 1);
    const int jc = min(max(jn, 0), MC - 1);
    const bool inr = (jn == jc);
    float u0 = hb[3 * jc + 0], u1 = hb[3 * jc + 1], u2 = hb[3 * jc + 2];
    nb0 = inr ? u0 : NEG2;
    nb1 = inr ? u1 : NEG2;
    nb2 = inr ? u2 : NEG2;
  }

  const float gx0 = gbuf[cur][j0] * LOG2E;
  const float gx1 = gbuf[cur][j1] * LOG2E;

  // Range gates: one sub + unsigned compares against precomputed spans.
  const unsigned u0g = (unsigned)(t - tlo0);
  const unsigned u1g = (unsigned)(t - tlo1);

  // Final-reduction contribution of diagonal t-2: x[p+1,q+1] there == g[t][j].
  if (u0g - 2u <= (unsigned)span0)
    merge3_2(h2.a[0] + gx0, h2.a[1] + gx0, h2.a[2] + gx0, s.rmx, s.rsm);
  if (u1g - 2u <= (unsigned)span1)
    merge3_2(h2.b[0] + gx1, h2.b[1] + gx1, h2.b[2] + gx1, s.rmx, s.rsm);

  // DP update. Cells outside their valid t-range are exactly -1e30 in the
  // fp32 reference recurrence (ulp(-1e30) >> all lse constants), so skip.
  float a0 = NEG2, r0 = NEG2, d0 = NEG2;
  if (u0g <= (unsigned)span0) {
    a0 = gx0 + lse4z_2(h2.a[0], h2.a[1], h2.a[2]);
    const float rsA = EVEN ? nb0 : h1.a[0];
    const float rsB = EVEN ? nb1 : h1.a[1];
    const float dsA = EVEN ? h1.a[0] : h1.b[0];
    const float dsB = EVEN ? h1.a[1] : h1.b[1];
    const float dsC = EVEN ? h1.a[2] : h1.b[2];
    r0 = lse2_2(rsA + GO2, rsB + GE2);
    d0 = lse3_2(dsA + GO2, dsB + GO2, dsC + GE2);
  }
  float a1 = NEG2, r1 = NEG2, d1 = NEG2;
  if (u1g <= (unsigned)span1) {
    a1 = gx1 + lse4z_2(h2.b[0], h2.b[1], h2.b[2]);
    const float rsA = EVEN ? h1.a[0] : h1.b[0];
    const float rsB = EVEN ? h1.a[1] : h1.b[1];
    const float dsA = EVEN ? h1.b[0] : nb0;
    const float dsB = EVEN ? h1.b[1] : nb1;
    const float dsC = EVEN ? h1.b[2] : nb2;
    r1 = lse2_2(rsA + GO2, rsB + GE2);
    d1 = lse3_2(dsA + GO2, dsB + GO2, dsC + GE2);
  }

  // h(t) overwrites the dead h(t-2) set (renaming, no copies); publish; barrier.
  h2.a[0] = a0; h2.a[1] = r0; h2.a[2] = d0;
  h2.b[0] = a1; h2.b[1] = r1; h2.b[2] = d1;
  float* hw = hbuf[cur];
  hw[3 * j0 + 0] = a0; hw[3 * j0 + 1] = r0; hw[3 * j0 + 2] = d0;
  if (has1) {
    hw[3 * j1 + 0] = a1; hw[3 * j1 + 1] = r1; hw[3 * j1 + 2] = d1;
  }
  __syncthreads();
}

__global__ __launch_bounds__(1024, 1)
void sw_wavefront_kernel(const float* __restrict__ x, float* __restrict__ out) {
  // h double buffer: 2 x 2047 cells x 3 states (49,128 B)
  __shared__ __align__(16) float hbuf[2][MC * 3];
  // diagonal ring: g[t] in gbuf[t&1] (16,384 B); reused as reduction scratch
  __shared__ __align__(16) float gbuf[2][NX];

  const int  tid  = threadIdx.x;
  const int  j0   = 2 * tid;
  const int  j1   = j0 + 1;
  const bool has1 = (j1 < MC);

  // Contiguous valid-t range per cell (diamond support of the scan).
  int tlo0 = max(2 * j0 - 2046, 2045 - 2 * j0);
  int span0 = min(2 * j0 + 2047, 6138 - 2 * j0) - tlo0;
  int tlo1 = max(2 * j1 - 2046, 2045 - 2 * j1);
  int span1 = min(2 * j1 + 2047, 6138 - 2 * j1) - tlo1;
  if (span1 < 0) { tlo1 = 0x40000000; span1 = 0; }   // empty range (j==2047)

  HPair P, Q;          // alternating generation register sets
  Acc   s;
#pragma unroll
  for (int k = 0; k < 3; ++k) {
    P.a[k] = NEG2; P.b[k] = NEG2; Q.a[k] = NEG2; Q.b[k] = NEG2;
  }
  s.rmx = -3.0e38f;
  s.rsm = 0.0f;
  s.L0 = (1023 - j0) * 2047;   // raw linear index of diagonal 0, cell j0
  s.L1 = (1023 - j1) * 2047;

  // Prefill h(t=-1) buffer (read at t=0) and async-prefetch diagonal 0.
  hbuf[1][3 * j0 + 0] = NEG2; hbuf[1][3 * j0 + 1] = NEG2; hbuf[1][3 * j0 + 2] = NEG2;
  if (has1) {
    hbuf[1][3 * j1 + 0] = NEG2; hbuf[1][3 * j1 + 1] = NEG2; hbuf[1][3 * j1 + 2] = NEG2;
  }
  {
    int c0 = min(max(s.L0, 0), NX * NX - 1);
    int c1 = min(max(s.L1, 0), NX * NX - 1);
    async_gather_f32(x + c0, &gbuf[0][j0]);
    async_gather_f32(x + c1, &gbuf[0][j1]);
    s.L0 += 1;   // advance to diagonal 1 (diag 0 is even: +1)
    s.L1 += 1;
  }
  __syncthreads();

  // Entering even step t: P = h(t-1), Q = h(t-2); step writes h(t) over its
  // h2 argument, so swapping (P,Q)/(Q,P) performs the generation shift free.
  for (int t = 0; t + 1 < ND; t += 2) {
    sw_step<true >(t,     j0, has1, x, hbuf, gbuf, tlo0, span0, tlo1, span1, P, Q, s);
    sw_step<false>(t + 1, j0, has1, x, hbuf, gbuf, tlo0, span0, tlo1, span1, Q, P, s);
  }
  sw_step<true>(ND - 1, j0, has1, x, hbuf, gbuf, tlo0, span0, tlo1, span1, P, Q, s);
  // Now: P = h(ND-2), Q = h(ND-1).

  // Pending reduction contributions of the last two diagonals.
  {
    const int tA = ND - 2, midA = (2046 + tA) >> 1;   // uses P
    if ((unsigned)(tA - tlo0) <= (unsigned)span0) {
      int p = midA - j0, q = tA - p;
      float xf = x[(p + 1) * NX + (q + 1)] * LOG2E;
      merge3_2(P.a[0] + xf, P.a[1] + xf, P.a[2] + xf, s.rmx, s.rsm);
    }
    if ((unsigned)(tA - tlo1) <= (unsigned)span1) {
      int p = midA - j1, q = tA - p;
      float xf = x[(p + 1) * NX + (q + 1)] * LOG2E;
      merge3_2(P.b[0] + xf, P.b[1] + xf, P.b[2] + xf, s.rmx, s.rsm);
    }
    const int tB = ND - 1, midB = (2046 + tB) >> 1;   // uses Q
    if ((unsigned)(tB - tlo0) <= (unsigned)span0) {
      int p = midB - j0, q = tB - p;
      float xf = x[(p + 1) * NX + (q + 1)] * LOG2E;
      merge3_2(Q.a[0] + xf, Q.a[1] + xf, Q.a[2] + xf, s.rmx, s.rsm);
    }
    if ((unsigned)(tB - tlo1) <= (unsigned)span1) {
      int p = midB - j1, q = tB - p;
      float xf = x[(p + 1) * NX + (q + 1)] * LOG2E;
      merge3_2(Q.b[0] + xf, Q.b[1] + xf, Q.b[2] + xf, s.rmx, s.rsm);
    }
  }

  // Block-wide (max, sum2) reduction; reuse gbuf as scratch.
  WAIT_ASYNCCNT(0);   // drain final (unused) prefetch before LDS reuse
  __syncthreads();
  float2* red = reinterpret_cast<float2*>(&gbuf[0][0]);   // 1024 * 8 B
  red[tid].x = s.rmx;
  red[tid].y = s.rsm;
  __syncthreads();
  for (int off = 512; off > 0; off >>= 1) {
    if (tid < off) {
      float2 a = red[tid];
      float2 b = red[tid + off];
      float m  = fmaxf(a.x, b.x);
      float ss = a.y * ex2(a.x - m) + b.y * ex2(b.x - m);
      red[tid].x = m;
      red[tid].y = ss;
    }
    __syncthreads();
  }
  if (tid == 0) out[0] = (red[0].x + lg2(red[0].y)) * LN2;   // back to natural log
}

extern "C" void kernel_launch(void* const* d_in, const int* in_sizes, int n_in,
                              void* d_out, int out_size, void* d_ws, size_t ws_size,
                              hipStream_t stream) {
  (void)in_sizes; (void)n_in; (void)out_size; (void)d_ws; (void)ws_size;
  const float* x = (const float*)d_in[0];
  float* out = (float*)d_out;
  // Sequential wavefront: one workgroup (32 wave32 waves) owns the whole DP.
  sw_wavefront_kernel<<<dim3(1), dim3(1024), 0, stream>>>(x, out);
}